// CombinedLoss_8572754723319
// MI455X (gfx1250) — compile-verified
//
#include <hip/hip_runtime.h>
#include <hip/hip_bf16.h>

// ---------------------------------------------------------------------------
// Problem constants (from reference): B=64, M=80, T_OUT=2000, T_IN=400
// ---------------------------------------------------------------------------
#define BB      64
#define MM      80
#define T_OUT   2000
#define T_IN    400
#define N_MEL   (BB * MM * T_OUT)      // 10,240,000
#define N_GATE  (BB * T_OUT)           // 128,000
#define N_ALN   (BB * T_OUT * T_IN)    // 51,200,000

#define BLK     256
#define NB_MEL  4096
#define NB_GATE 128
#define NB_ATT  4096

typedef float f4  __attribute__((ext_vector_type(4)));
typedef float v2f __attribute__((ext_vector_type(2)));
typedef float v8f __attribute__((ext_vector_type(8)));

// ---------------------------------------------------------------------------
// Wave32 reduction via V_WMMA_F32_16X16X4_F32:
//   A (16x4, 2 VGPRs): a.x = per-lane value, a.y = 0
//     -> A[m,0] = v[lane m], A[m,2] = v[lane m+16], others 0
//   B (4x16) = ones  ->  D[m,n] = v[m] + v[m+16]  (all n)
//   D layout: VGPR r holds row r (lanes 0-15) and row r+8 (lanes 16-31),
//   so sum(d[0..7]) gives each half-wave's half-sum; one xor-16 shuffle
//   completes the 32-lane sum. EXEC must be all-ones at the call site.
// ---------------------------------------------------------------------------
__device__ __forceinline__ float wave_reduce_wmma(float v) {
    v2f a;    a.x = v;    a.y = 0.0f;
    v2f ones; ones.x = 1.0f; ones.y = 1.0f;
    v8f c = {};
    v8f d = __builtin_amdgcn_wmma_f32_16x16x4_f32(
        /*neg_a=*/false, a, /*neg_b=*/false, ones,
        /*c_mod=*/(short)0, c, /*reuse_a=*/false, /*reuse_b=*/false);
    float h = d[0] + d[1] + d[2] + d[3] + d[4] + d[5] + d[6] + d[7];
    h += __shfl_xor(h, 16, 32);   // combine the two half-wave sums
    return h;                      // full 32-lane sum, in every lane
}

// Deterministic block reduction (BLK threads). Result valid on thread 0.
__device__ __forceinline__ float block_reduce(float v) {
    __shared__ float smem[BLK / 32];
    float w = wave_reduce_wmma(v);
    const unsigned lane = threadIdx.x & 31u;
    const unsigned wid  = threadIdx.x >> 5;
    if (lane == 0) smem[wid] = w;
    __syncthreads();
    float r = 0.0f;
    if (threadIdx.x == 0) {
        #pragma unroll
        for (int k = 0; k < BLK / 32; ++k) r += smem[k];
    }
    __syncthreads();   // make smem reusable for back-to-back reductions
    return r;
}

// ---------------------------------------------------------------------------
// Kernel 1: mel L1 partials.  sum(|mo-t|) + sum(|mp-t|), 128-bit NT loads.
// ---------------------------------------------------------------------------
__global__ void mel_l1_kernel(const float* __restrict__ mo,
                              const float* __restrict__ mp,
                              const float* __restrict__ mt,
                              float* __restrict__ part) {
    const int n4 = N_MEL / 4;
    const int stride = gridDim.x * blockDim.x;
    float acc = 0.0f;
    for (int idx = blockIdx.x * blockDim.x + threadIdx.x; idx < n4; idx += stride) {
        f4 a = __builtin_nontemporal_load((const f4*)mo + idx);
        f4 b = __builtin_nontemporal_load((const f4*)mp + idx);
        f4 t = __builtin_nontemporal_load((const f4*)mt + idx);
        acc += fabsf(a.x - t.x) + fabsf(a.y - t.y) + fabsf(a.z - t.z) + fabsf(a.w - t.w);
        acc += fabsf(b.x - t.x) + fabsf(b.y - t.y) + fabsf(b.z - t.z) + fabsf(b.w - t.w);
    }
    float s = block_reduce(acc);
    if (threadIdx.x == 0) part[blockIdx.x] = s;
}

// ---------------------------------------------------------------------------
// Kernel 2: gate BCE-with-logits partials.
//   term = max(x,0) - x*z + log1p(exp(-|x|))
// ---------------------------------------------------------------------------
__global__ void gate_bce_kernel(const float* __restrict__ go,
                                const float* __restrict__ gt,
                                float* __restrict__ part) {
    const int n4 = N_GATE / 4;
    const int stride = gridDim.x * blockDim.x;
    float acc = 0.0f;
    for (int idx = blockIdx.x * blockDim.x + threadIdx.x; idx < n4; idx += stride) {
        f4 x = __builtin_nontemporal_load((const f4*)go + idx);
        f4 z = __builtin_nontemporal_load((const f4*)gt + idx);
        #pragma unroll
        for (int c = 0; c < 4; ++c) {
            float xc = x[c], zc = z[c];
            acc += fmaxf(xc, 0.0f) - xc * zc + log1pf(__expf(-fabsf(xc)));
        }
    }
    float s = block_reduce(acc);
    if (threadIdx.x == 0) part[blockIdx.x] = s;
}

// ---------------------------------------------------------------------------
// Kernel 3: single pass over alignments computing BOTH attention sums.
//   att += a[b,i,j] * (j < i)
//   ga  += a[b,i,j] * (1 - exp(-(i - j*ol/il)^2 * 3.125)) for i<ol, j<il
// T_IN % 4 == 0, so a float4 never crosses a (b,i) row.
// ---------------------------------------------------------------------------
__global__ void att_kernel(const float* __restrict__ aln,
                           const int* __restrict__ in_len,
                           const int* __restrict__ out_len,
                           float* __restrict__ att_part,
                           float* __restrict__ ga_part) {
    const int n4    = N_ALN / 4;            // 12,800,000
    const int ROW4  = T_IN / 4;             // 100 float4 per row
    const int PERB4 = (T_OUT * T_IN) / 4;   // 200,000 float4 per batch
    const int stride = gridDim.x * blockDim.x;
    float att = 0.0f, ga = 0.0f;
    for (int idx = blockIdx.x * blockDim.x + threadIdx.x; idx < n4; idx += stride) {
        f4 a = __builtin_nontemporal_load((const f4*)aln + idx);
        int b  = idx / PERB4;
        int r  = idx - b * PERB4;
        int i  = r / ROW4;
        int j0 = (r - i * ROW4) * 4;
        int il = in_len[b];
        int ol = out_len[b];
        float ratio = (float)ol / (float)il;
        float fi = (float)i;
        bool  vi = (i < ol);
        #pragma unroll
        for (int c = 0; c < 4; ++c) {
            float av = a[c];
            int   j  = j0 + c;
            if (j < i) att += av;
            if (vi && (j < il)) {
                float dd = fi - (float)j * ratio;
                float w  = 1.0f - __expf(-dd * dd * 3.125f);  // 1/(2*0.4^2)=3.125
                ga += av * w;
            }
        }
    }
    float s0 = block_reduce(att);
    if (threadIdx.x == 0) att_part[blockIdx.x] = s0;
    float s1 = block_reduce(ga);
    if (threadIdx.x == 0) ga_part[blockIdx.x] = s1;
}

// ---------------------------------------------------------------------------
// Kernel 4: deterministic finalize — reduce partial arrays, emit 5 outputs.
// ---------------------------------------------------------------------------
__global__ void finalize_kernel(const float* __restrict__ ws,
                                float* __restrict__ out) {
    const float* mel_p  = ws;
    const float* gate_p = mel_p  + NB_MEL;
    const float* att_p  = gate_p + NB_GATE;
    const float* ga_p   = att_p  + NB_ATT;
    float a = 0.0f, b = 0.0f, c = 0.0f, d = 0.0f;
    for (int k = threadIdx.x; k < NB_MEL;  k += BLK) a += mel_p[k];
    for (int k = threadIdx.x; k < NB_GATE; k += BLK) b += gate_p[k];
    for (int k = threadIdx.x; k < NB_ATT;  k += BLK) c += att_p[k];
    for (int k = threadIdx.x; k < NB_ATT;  k += BLK) d += ga_p[k];
    float mel  = block_reduce(a);
    float gate = block_reduce(b);
    float att  = block_reduce(c);
    float ga   = block_reduce(d);
    if (threadIdx.x == 0) {
        mel  /= (float)N_MEL;
        gate /= (float)N_GATE;
        att  /= (float)BB;
        ga   /= (float)BB;
        out[0] = mel + gate + 0.1f * att + 0.1f * ga;  // total
        out[1] = mel;
        out[2] = gate;
        out[3] = att;
        out[4] = ga;
    }
}

// ---------------------------------------------------------------------------
// Launcher
// ---------------------------------------------------------------------------
extern "C" void kernel_launch(void* const* d_in, const int* in_sizes, int n_in,
                              void* d_out, int out_size, void* d_ws, size_t ws_size,
                              hipStream_t stream) {
    (void)in_sizes; (void)n_in; (void)out_size; (void)ws_size;
    const float* mel_out         = (const float*)d_in[0];
    const float* mel_out_postnet = (const float*)d_in[1];
    const float* gate_out        = (const float*)d_in[2];
    const float* alignments      = (const float*)d_in[3];
    const float* mel_target      = (const float*)d_in[4];
    const float* gate_target     = (const float*)d_in[5];
    const int*   input_lengths   = (const int*)d_in[6];
    const int*   output_lengths  = (const int*)d_in[7];
    float* out = (float*)d_out;

    float* ws     = (float*)d_ws;
    float* mel_p  = ws;
    float* gate_p = mel_p  + NB_MEL;
    float* att_p  = gate_p + NB_GATE;
    float* ga_p   = att_p  + NB_ATT;

    mel_l1_kernel  <<<NB_MEL,  BLK, 0, stream>>>(mel_out, mel_out_postnet, mel_target, mel_p);
    gate_bce_kernel<<<NB_GATE, BLK, 0, stream>>>(gate_out, gate_target, gate_p);
    att_kernel     <<<NB_ATT,  BLK, 0, stream>>>(alignments, input_lengths, output_lengths,
                                                 att_p, ga_p);
    finalize_kernel<<<1,       BLK, 0, stream>>>(ws, out);
}